// InvAttention_concat_67156108640593
// MI455X (gfx1250) — compile-verified
//
#include <hip/hip_runtime.h>
#include <math.h>

// InvAttention_concat on MI455X (gfx1250, wave32).
// Attention collapses to rank-1 structure; VJP is a rank-4 update per batch.
// f32 WMMA (16x16x4) for GEMM-shaped pieces; b128-vectorized streaming for the rest.

#define LAM  1.0000001f
#define NPOS 1024
#define CCH  256
#define NB   8

typedef float v2f __attribute__((ext_vector_type(2)));
typedef float v4f __attribute__((ext_vector_type(4)));
typedef float v8f __attribute__((ext_vector_type(8)));

__device__ __forceinline__ float wave_sum(float v) {
#pragma unroll
  for (int off = 16; off > 0; off >>= 1) v += __shfl_xor(v, off, 32);
  return v;
}

// deterministic block reduction
__device__ __forceinline__ float block_sum(float v, float* sm) {
  int t = threadIdx.x;
  v = wave_sum(v);
  if ((t & 31) == 0) sm[t >> 5] = v;
  __syncthreads();
  if (t < 32) {
    float x = (t < ((int)blockDim.x >> 5)) ? sm[t] : 0.0f;
    x = wave_sum(x);
    if (t == 0) sm[0] = x;
  }
  __syncthreads();
  float r = sm[0];
  __syncthreads();
  return r;
}

// ---- K0: a = Wq^T wc_q, b = Wk^T wc_k (into padded 16x256 RHS buffer), scalars, zero tr
__global__ __launch_bounds__(256) void k_prep(const float* __restrict__ Wq, const float* __restrict__ bq,
                                              const float* __restrict__ Wk, const float* __restrict__ bk,
                                              const float* __restrict__ Wc,
                                              float* __restrict__ G16, float* __restrict__ SCAL,
                                              float* __restrict__ TRv) {
  int d = threadIdx.x;
  float av = 0.f, bvv = 0.f;
  for (int o = 0; o < 64; ++o) {
    av  += Wc[o]      * Wq[o * 256 + d];
    bvv += Wc[64 + o] * Wk[o * 256 + d];
  }
  G16[d]       = av;   // col 0 = a
  G16[256 + d] = bvv;  // col 1 = b
  for (int k = 2; k < 16; ++k) G16[k * 256 + d] = 0.f;  // zero padding cols
  if (d == 0) {
    float ca = 0.f, cb = 0.f;
    for (int o = 0; o < 64; ++o) { ca += Wc[o] * bq[o]; cb += Wc[64 + o] * bk[o]; }
    SCAL[0] = ca; SCAL[1] = cb;
  }
  if (d < 8) TRv[d] = 0.f;
}

// ---- K1 (WMMA): per batch  Y[n, 0:16] = X_b^T * G16  -> p (col0), r (col1)
__global__ __launch_bounds__(32) void k_xt_ab(const float* __restrict__ X, const float* __restrict__ G16,
                                              float* __restrict__ P, float* __restrict__ R) {
  int b    = blockIdx.y;
  int n0   = blockIdx.x << 4;
  int lane = threadIdx.x;
  int half = lane >> 4, l15 = lane & 15;
  const float* Xb = X + (size_t)b * CCH * NPOS;
  v8f c = {};
  for (int d0 = 0; d0 < CCH; d0 += 4) {
    int dA = d0 + half * 2;
    v2f a, bb;
    a.x  = Xb[dA * NPOS + n0 + l15];        // A(M=l15, K=dA-d0)
    a.y  = Xb[(dA + 1) * NPOS + n0 + l15];
    bb.x = G16[l15 * 256 + dA];             // B(K, N=l15)
    bb.y = G16[l15 * 256 + dA + 1];
    c = __builtin_amdgcn_wmma_f32_16x16x4_f32(false, a, false, bb, (short)0, c, false, false);
  }
  int rowbase = n0 + half * 8;              // D: lane=col, VGPR=row (+8 upper half)
  if (l15 == 0) {
    for (int v = 0; v < 8; ++v) P[b * NPOS + rowbase + v] = c[v];
  } else if (l15 == 1) {
    for (int v = 0; v < 8; ++v) R[b * NPOS + rowbase + v] = c[v];
  }
}

// ---- shared WMMA GEMM: OUT[256,16] = Wv (opt transposed) * RHS[16 cols][256]
__global__ __launch_bounds__(32) void k_gemm256(const float* __restrict__ Wv, const float* __restrict__ RHS,
                                                float* __restrict__ OUT, int transA) {
  int d0   = blockIdx.x << 4;
  int lane = threadIdx.x;
  int half = lane >> 4, l15 = lane & 15;
  v8f c = {};
  for (int k0 = 0; k0 < 256; k0 += 4) {
    int kk = k0 + half * 2;
    v2f a, bb;
    if (transA) {                           // A(m,k) = Wv[k][m]  (coalesced over lanes)
      a.x = Wv[kk * 256 + d0 + l15];
      a.y = Wv[(kk + 1) * 256 + d0 + l15];
    } else {                                // A(m,k) = Wv[m][k]
      a.x = Wv[(d0 + l15) * 256 + kk];
      a.y = Wv[(d0 + l15) * 256 + kk + 1];
    }
    bb.x = RHS[l15 * 256 + kk];
    bb.y = RHS[l15 * 256 + kk + 1];
    c = __builtin_amdgcn_wmma_f32_16x16x4_f32(false, a, false, bb, (short)0, c, false, false);
  }
  for (int v = 0; v < 8; ++v) OUT[(d0 + half * 8 + v) * 16 + l15] = c[v];
}

// ---- K2: per (b,c): (X p)[c] and row-sum xs[c] -> RHS1 cols [0:8]=Xp, [8:16]=xs
__global__ __launch_bounds__(256) void k_rowstats(const float* __restrict__ X, const float* __restrict__ P,
                                                  float* __restrict__ RHS1) {
  int b = blockIdx.x;
  int c = (blockIdx.y << 3) + (threadIdx.x >> 5);
  int lane = threadIdx.x & 31;
  const float4* xr = (const float4*)(X + ((size_t)b * CCH + c) * NPOS);
  const float4* pb = (const float4*)(P + b * NPOS);
  float sxp = 0.f, sx = 0.f;
#pragma unroll
  for (int q = 0; q < 8; ++q) {
    int i = q * 32 + lane;
    float4 xv = xr[i], pv = pb[i];
    sxp += xv.x * pv.x + xv.y * pv.y + xv.z * pv.z + xv.w * pv.w;
    sx  += xv.x + xv.y + xv.z + xv.w;
  }
  sxp = wave_sum(sxp); sx = wave_sum(sx);
  if (lane == 0) { RHS1[b * 256 + c] = sxp; RHS1[(8 + b) * 256 + c] = sx; }
}

__global__ __launch_bounds__(256) void k_sump(const float* __restrict__ P, float* __restrict__ SUMP) {
  __shared__ float sm[32];
  int b = blockIdx.x;
  float4 pv = ((const float4*)(P + b * NPOS))[threadIdx.x];
  float s = block_sum(pv.x + pv.y + pv.z + pv.w, sm);
  if (threadIdx.x == 0) SUMP[b] = s;
}

// ---- K4: s1/s2 for x and for x*LAM (analytic scaling of bias-free vs bias parts)
__global__ __launch_bounds__(256) void k_fold(const float* __restrict__ A12, const float* __restrict__ bv,
                                              const float* __restrict__ SUMP, const float* __restrict__ SCAL,
                                              float* __restrict__ S1, float* __restrict__ S2,
                                              float* __restrict__ S1P, float* __restrict__ S2P) {
  int b = blockIdx.x, d = threadIdx.x;
  float A1 = A12[d * 16 + b], A2 = A12[d * 16 + 8 + b];
  float ca = SCAL[0];
  float bvd = bv[d], sp = SUMP[b];
  float lam = LAM, lam2 = LAM * LAM;
  S1[b * 256 + d]  = A1 + ca * A2 + bvd * (sp + 1024.f * ca);
  S2[b * 256 + d]  = A2 + 1024.f * bvd;
  S1P[b * 256 + d] = lam2 * A1 + lam * ca * A2 + bvd * (lam * sp + 1024.f * ca);
  S2P[b * 256 + d] = lam * A2 + 1024.f * bvd;
}

// ---- K5: ||att(x*LAM)-att(x)||^2 and ||x*LAM-x||^2 partials; one block per (b,d) row
__global__ __launch_bounds__(256) void k_lipnorm(const float* __restrict__ X, const float* __restrict__ R,
                                                 const float* __restrict__ S1, const float* __restrict__ S2,
                                                 const float* __restrict__ S1P, const float* __restrict__ S2P,
                                                 const float* __restrict__ SCAL, const float* __restrict__ gamma,
                                                 float* __restrict__ PNUM, float* __restrict__ PDEN) {
  __shared__ float sm[32];
  int b = blockIdx.x >> 8, d = blockIdx.x & 255;
  float cb  = SCAL[1];
  float g   = gamma[0] * (1.0f / 1024.f);
  float s1v = S1[b * 256 + d],  s2v = S2[b * 256 + d];
  float s1p = S1P[b * 256 + d], s2p = S2P[b * 256 + d];
  float4 xv = ((const float4*)X)[(size_t)blockIdx.x * 256 + threadIdx.x];
  float4 rv = ((const float4*)(R + b * NPOS))[threadIdx.x];
  float an = 0.f, ad = 0.f;
  {
    float xs[4] = {xv.x, xv.y, xv.z, xv.w};
    float rs[4] = {rv.x, rv.y, rv.z, rv.w};
#pragma unroll
    for (int q = 0; q < 4; ++q) {
      float a1 = xs[q] + g * (s1v + (rs[q] + cb) * s2v);
      float a2 = LAM * xs[q] + g * (s1p + (LAM * rs[q] + cb) * s2p);
      float dn = a2 - a1;
      float dd = LAM * xs[q] - xs[q];
      an += dn * dn; ad += dd * dd;
    }
  }
  an = block_sum(an, sm);
  ad = block_sum(ad, sm);
  if (threadIdx.x == 0) { PNUM[blockIdx.x] = an; PDEN[blockIdx.x] = ad; }
}

__global__ __launch_bounds__(256) void k_lipfinal(const float* __restrict__ PNUM, const float* __restrict__ PDEN,
                                                  const float* __restrict__ gamma, float* __restrict__ SCAL) {
  __shared__ float sm[32];
  float n = 0.f, d = 0.f;
  for (int i = threadIdx.x; i < 2048; i += 256) { n += PNUM[i]; d += PDEN[i]; }
  n = block_sum(n, sm);
  d = block_sum(d, sm);
  if (threadIdx.x == 0) {
    float lip = sqrtf(n) / sqrtf(d);
    float factor = (lip > 0.9f) ? (0.9f / lip) : 1.0f;
    SCAL[2] = gamma[0] * factor;   // gamma_eff
  }
}

// ---- K7: out = x + Fx = 2x + gamma_eff * out(x); nontemporal b128 stores
__global__ __launch_bounds__(256) void k_final(const float* __restrict__ X, const float* __restrict__ R,
                                               const float* __restrict__ S1, const float* __restrict__ S2,
                                               const float* __restrict__ SCAL, float* __restrict__ OUT) {
  int b = blockIdx.x >> 8, d = blockIdx.x & 255;
  float cb  = SCAL[1];
  float ge  = SCAL[2] * (1.0f / 1024.f);
  float s1v = S1[b * 256 + d], s2v = S2[b * 256 + d];
  float4 xv = ((const float4*)X)[(size_t)blockIdx.x * 256 + threadIdx.x];
  float4 rv = ((const float4*)(R + b * NPOS))[threadIdx.x];
  v4f o;
  o.x = 2.f * xv.x + ge * (s1v + (rv.x + cb) * s2v);
  o.y = 2.f * xv.y + ge * (s1v + (rv.y + cb) * s2v);
  o.z = 2.f * xv.z + ge * (s1v + (rv.z + cb) * s2v);
  o.w = 2.f * xv.w + ge * (s1v + (rv.w + cb) * s2v);
  __builtin_nontemporal_store(o, (v4f*)OUT + (size_t)blockIdx.x * 256 + threadIdx.x);
}

// ---- K8: per (b,c): rowsum(w) and sum_m w*ek -> RHS2 cols [0:8]=Wsum, [8:16]=we
__global__ __launch_bounds__(256) void k_wrowstats(const float* __restrict__ W, const float* __restrict__ R,
                                                   const float* __restrict__ SCAL, float* __restrict__ RHS2) {
  int b = blockIdx.x;
  int c = (blockIdx.y << 3) + (threadIdx.x >> 5);
  int lane = threadIdx.x & 31;
  float cb = SCAL[1];
  const float4* wr = (const float4*)(W + ((size_t)b * CCH + c) * NPOS);
  const float4* rb = (const float4*)(R + b * NPOS);
  float sw = 0.f, swe = 0.f;
#pragma unroll
  for (int q = 0; q < 8; ++q) {
    int i = q * 32 + lane;
    float4 wv = wr[i], rv = rb[i];
    sw  += wv.x + wv.y + wv.z + wv.w;
    swe += wv.x * (rv.x + cb) + wv.y * (rv.y + cb) + wv.z * (rv.z + cb) + wv.w * (rv.w + cb);
  }
  sw = wave_sum(sw); swe = wave_sum(swe);
  if (lane == 0) { RHS2[b * 256 + c] = sw; RHS2[(8 + b) * 256 + c] = swe; }
}

__global__ __launch_bounds__(256) void k_bvdot(const float* __restrict__ bv, const float* __restrict__ RHS2,
                                               float* __restrict__ BVDOT) {
  int b = threadIdx.x >> 5;
  int lane = threadIdx.x & 31;
  float s = 0.f;
  for (int c = lane; c < 256; c += 32) s += bv[c] * RHS2[b * 256 + c];
  s = wave_sum(s);
  if (lane == 0) BVDOT[b] = s;
}

// ---- K11: h1[n] = X^T g1 + bv.Wsum ; h2[n] = w^T s2   (block: b = x, n-chunk = y)
__global__ __launch_bounds__(64) void k_hvec(const float* __restrict__ X, const float* __restrict__ W,
                                             const float* __restrict__ G12, const float* __restrict__ S2,
                                             const float* __restrict__ BVDOT,
                                             float* __restrict__ H1, float* __restrict__ H2) {
  int b  = blockIdx.x;
  int i4 = (blockIdx.y << 6) + threadIdx.x;      // float4 index within the 1024-row
  const float4* Xb = (const float4*)(X + (size_t)b * CCH * NPOS);
  const float4* Wb = (const float4*)(W + (size_t)b * CCH * NPOS);
  float4 h1 = {0.f, 0.f, 0.f, 0.f}, h2 = {0.f, 0.f, 0.f, 0.f};
  for (int d = 0; d < CCH; ++d) {
    float g1 = G12[d * 16 + b];
    float s2 = S2[b * 256 + d];
    float4 xv = Xb[d * 256 + i4];
    float4 wv = Wb[d * 256 + i4];
    h1.x += xv.x * g1; h1.y += xv.y * g1; h1.z += xv.z * g1; h1.w += xv.w * g1;
    h2.x += wv.x * s2; h2.y += wv.y * s2; h2.z += wv.z * s2; h2.w += wv.w * s2;
  }
  float bd = BVDOT[b];
  h1.x += bd; h1.y += bd; h1.z += bd; h1.w += bd;
  ((float4*)(H1 + b * NPOS))[i4] = h1;
  ((float4*)(H2 + b * NPOS))[i4] = h2;
}

// ---- K12: w' = w + (ge/N)(g1*eq + a*h1 + g2 + b*h2) in place; partial dot(w',u)
__global__ __launch_bounds__(256) void k_update(float* __restrict__ W, const float* __restrict__ U,
                                                const float* __restrict__ P, const float* __restrict__ H1,
                                                const float* __restrict__ H2, const float* __restrict__ G12,
                                                const float* __restrict__ G16, const float* __restrict__ SCAL,
                                                float* __restrict__ DOTP) {
  __shared__ float sm[32];
  int b = blockIdx.y, d = blockIdx.x;
  float geN = SCAL[2] * (1.0f / 1024.f);
  float ca  = SCAL[0];
  float g1  = G12[d * 16 + b], g2 = G12[d * 16 + 8 + b];
  float av  = G16[d], bvv = G16[256 + d];
  size_t off4 = ((size_t)b * CCH + d) * 256;     // float4 units
  int t = threadIdx.x;
  float4 wv = ((float4*)W)[off4 + t];
  float4 uv = ((const float4*)U)[off4 + t];
  float4 pv = ((const float4*)(P + b * NPOS))[t];
  float4 h1 = ((const float4*)(H1 + b * NPOS))[t];
  float4 h2 = ((const float4*)(H2 + b * NPOS))[t];
  float4 o;
  o.x = wv.x + geN * (g1 * (pv.x + ca) + av * h1.x + g2 + bvv * h2.x);
  o.y = wv.y + geN * (g1 * (pv.y + ca) + av * h1.y + g2 + bvv * h2.y);
  o.z = wv.z + geN * (g1 * (pv.z + ca) + av * h1.z + g2 + bvv * h2.z);
  o.w = wv.w + geN * (g1 * (pv.w + ca) + av * h1.w + g2 + bvv * h2.w);
  ((float4*)W)[off4 + t] = o;
  float acc = o.x * uv.x + o.y * uv.y + o.z * uv.z + o.w * uv.w;
  acc = block_sum(acc, sm);
  if (t == 0) DOTP[b * 256 + d] = acc;
}

__global__ __launch_bounds__(256) void k_dotred(const float* __restrict__ DOTP, float* __restrict__ TRv,
                                                float* __restrict__ OUTTR, float coef) {
  __shared__ float sm[32];
  int b = blockIdx.x;
  float s = DOTP[b * 256 + threadIdx.x];
  s = block_sum(s, sm);
  if (threadIdx.x == 0) { float t = TRv[b] + coef * s; TRv[b] = t; OUTTR[b] = t; }
}

extern "C" void kernel_launch(void* const* d_in, const int* in_sizes, int n_in,
                              void* d_out, int out_size, void* d_ws, size_t ws_size,
                              hipStream_t stream) {
  const float* x     = (const float*)d_in[0];
  const float* Wq    = (const float*)d_in[1];
  const float* bq    = (const float*)d_in[2];
  const float* Wk    = (const float*)d_in[3];
  const float* bk    = (const float*)d_in[4];
  const float* Wc    = (const float*)d_in[5];
  const float* Wv    = (const float*)d_in[6];
  const float* bv    = (const float*)d_in[7];
  const float* gamma = (const float*)d_in[8];
  const float* u     = (const float*)d_in[9];
  float* out = (float*)d_out;
  float* ws  = (float*)d_ws;

  // workspace layout (floats); total ~2.17M floats (~8.7 MB); all offsets 16B-aligned
  float* G16   = ws;            // [16][256] padded RHS: col0=a, col1=b
  float* SCAL  = ws + 4096;     // [16] : 0=c_a 1=c_b 2=gamma_eff
  float* P     = ws + 4112;     // [8][1024] p = X^T a
  float* R     = ws + 12304;    // [8][1024] r = X^T b
  float* RHS1  = ws + 20496;    // [16][256] cols: Xp(8) | xs(8)
  float* A12   = ws + 24592;    // [256][16] cols: A1(8) | A2(8)
  float* S1    = ws + 28688;    // [8][256]
  float* S2    = ws + 30736;    // [8][256]
  float* S1P   = ws + 32784;    // [8][256]
  float* S2P   = ws + 34832;    // [8][256]
  float* SUMP  = ws + 36880;    // [8]
  float* BVDOT = ws + 36888;    // [8]
  float* TRv   = ws + 36896;    // [8]
  float* PNUM  = ws + 36904;    // [2048]
  float* PDEN  = ws + 38952;    // [2048]
  float* DOTP  = ws + 41000;    // [8][256]
  float* RHS2  = ws + 43048;    // [16][256] cols: Wsum(8) | we(8)
  float* G12   = ws + 47144;    // [256][16] cols: g1(8) | g2(8)
  float* H1    = ws + 51240;    // [8][1024]
  float* H2    = ws + 59432;    // [8][1024]
  float* WBUF  = ws + 67624;    // [8][256][1024] vjp cotangent (in-place updated)

  k_prep<<<1, 256, 0, stream>>>(Wq, bq, Wk, bk, Wc, G16, SCAL, TRv);
  (void)hipMemcpyAsync(WBUF, u, (size_t)NB * CCH * NPOS * sizeof(float),
                       hipMemcpyDeviceToDevice, stream);

  k_xt_ab<<<dim3(64, 8), 32, 0, stream>>>(x, G16, P, R);          // WMMA
  k_rowstats<<<dim3(8, 32), 256, 0, stream>>>(x, P, RHS1);
  k_sump<<<8, 256, 0, stream>>>(P, SUMP);
  k_gemm256<<<16, 32, 0, stream>>>(Wv, RHS1, A12, 0);             // WMMA
  k_fold<<<8, 256, 0, stream>>>(A12, bv, SUMP, SCAL, S1, S2, S1P, S2P);
  k_lipnorm<<<2048, 256, 0, stream>>>(x, R, S1, S2, S1P, S2P, SCAL, gamma, PNUM, PDEN);
  k_lipfinal<<<1, 256, 0, stream>>>(PNUM, PDEN, gamma, SCAL);
  k_final<<<2048, 256, 0, stream>>>(x, R, S1, S2, SCAL, out);

  const float coefs[5] = {1.f, -0.5f, 1.f / 3.f, -0.25f, 0.2f};
  for (int j = 0; j < 5; ++j) {
    k_wrowstats<<<dim3(8, 32), 256, 0, stream>>>(WBUF, R, SCAL, RHS2);
    k_gemm256<<<16, 32, 0, stream>>>(Wv, RHS2, G12, 1);           // WMMA (Wv^T)
    k_bvdot<<<1, 256, 0, stream>>>(bv, RHS2, BVDOT);
    k_hvec<<<dim3(8, 4), 64, 0, stream>>>(x, WBUF, G12, S2, BVDOT, H1, H2);
    k_update<<<dim3(256, 8), 256, 0, stream>>>(WBUF, u, P, H1, H2, G12, G16, SCAL, DOTP);
    k_dotred<<<8, 256, 0, stream>>>(DOTP, TRv, out + 2097152, coefs[j]);
  }

  (void)in_sizes; (void)n_in; (void)out_size; (void)ws_size;
}